// Lift2DTo3D_64355789963833
// MI455X (gfx1250) — compile-verified
//
#include <hip/hip_runtime.h>

// ---------------- problem constants (from the reference) ----------------
#define BT_   2                       // B*T
#define VC_   6
#define C_    256
#define H4_   64
#define W4_   112
#define HW_   (H4_ * W4_)             // 7168
#define NPT_  (VC_ * HW_)             // 43008 points per (b,t)
#define NZ_   10
#define NY_   100
#define NX_   100
#define NV_   (NZ_ * NY_ * NX_)       // 100000 voxels
#define LDS_STRIDE 260                // 256 + 4 pad -> conflict-free ds_load_b64
                                      // (260*4 = 1040 B, 16B aligned for b128)

typedef float v2f __attribute__((ext_vector_type(2)));
typedef float v8f __attribute__((ext_vector_type(8)));
typedef int   v4i_vs __attribute__((vector_size(4 * sizeof(int))));  // builtin's B128 unit

#if __has_builtin(__builtin_amdgcn_global_load_async_to_lds_b128)
#define ASYNC_STAGE 1
#else
#define ASYNC_STAGE 0
#endif

// ---------------------------------------------------------------------------
// Kernel 1: zero workspace (vol_feat + vol_w). Must run every launch because
// the scatter phase accumulates with atomics.
// ---------------------------------------------------------------------------
__global__ void zero_ws_kernel(float4* __restrict__ p, long long n4) {
  long long i = (long long)blockIdx.x * blockDim.x + threadIdx.x;
  const long long s = (long long)gridDim.x * blockDim.x;
  const float4 z = make_float4(0.f, 0.f, 0.f, 0.f);
  for (; i < n4; i += s) p[i] = z;
}

// ---------------------------------------------------------------------------
// Kernel 2: trilinear scatter of point features into the voxel grid.
// One thread per point; coalesced along the contiguous hw axis of feat/xyz.
// Faithfully replicates the reference's cumulative `valid` AND across the
// 8 corners in (dx, dy, dz) iteration order.
// ---------------------------------------------------------------------------
__global__ void lift_scatter_kernel(const float* __restrict__ feat,
                                    const float* __restrict__ xyz,
                                    const float* __restrict__ conf,
                                    float* __restrict__ vol_feat,   // [BT][NV][C]
                                    float* __restrict__ vol_w)      // [BT][NV]
{
  const int p = blockIdx.x * blockDim.x + threadIdx.x;
  if (p >= BT_ * NPT_) return;
  const int bt = p / NPT_;
  const int n  = p - bt * NPT_;
  const int vc = n / HW_;
  const int hw = n - vc * HW_;

  const float* xb = xyz + ((bt * VC_ + vc) * 3) * HW_ + hw;
  const float x  = xb[0];
  const float y  = xb[HW_];
  const float z  = xb[2 * HW_];
  const float cf = conf[(bt * VC_ + vc) * HW_ + hw];

  bool valid = __builtin_isfinite(x) && __builtin_isfinite(y) && __builtin_isfinite(z);
  valid = valid && (cf > 1e-4f);
  valid = valid && (x >= -40.f) && (x < 40.f);
  valid = valid && (y >= -40.f) && (y < 40.f);
  valid = valid && (z >=  -2.f) && (z <  6.f);
  if (!valid) return;   // if base-invalid, every corner weight is 0

  const float xi = (x + 40.f) / 0.8f;
  const float yi = (y + 40.f) / 0.8f;
  const float zi = (z +  2.f) / 0.8f;
  const float x0 = floorf(xi), y0 = floorf(yi), z0 = floorf(zi);
  const float fx = fminf(fmaxf(xi - x0, 0.f), 1.f);
  const float fy = fminf(fmaxf(yi - y0, 0.f), 1.f);
  const float fz = fminf(fmaxf(zi - z0, 0.f), 1.f);
  const int x0i = (int)x0, y0i = (int)y0, z0i = (int)z0;

  int   lin8[8];
  float w8[8];
  int m = 0;
  bool v = true;   // cumulative, matching the reference's aliased `valid`
#pragma unroll
  for (int dx = 0; dx < 2; ++dx)
#pragma unroll
    for (int dy = 0; dy < 2; ++dy)
#pragma unroll
      for (int dz = 0; dz < 2; ++dz) {
        const int jx = x0i + dx, jy = y0i + dy, jz = z0i + dz;
        v = v && (jx >= 0) && (jx < NX_) && (jy >= 0) && (jy < NY_)
              && (jz >= 0) && (jz < NZ_);
        const int ix = min(max(jx, 0), NX_ - 1);
        const int iy = min(max(jy, 0), NY_ - 1);
        const int iz = min(max(jz, 0), NZ_ - 1);
        const float wx = dx ? fx : 1.f - fx;
        const float wy = dy ? fy : 1.f - fy;
        const float wz = dz ? fz : 1.f - fz;
        const float w  = wx * wy * wz * cf * (v ? 1.f : 0.f);
        if (w != 0.f) {
          lin8[m] = iz * (NY_ * NX_) + iy * NX_ + ix;
          w8[m]   = w;
          ++m;
        }
      }
  if (m == 0) return;

  float* vw = vol_w + bt * NV_;
  float* vf = vol_feat + (long long)bt * NV_ * C_;
  for (int k = 0; k < m; ++k) atomicAdd(&vw[lin8[k]], w8[k]);

  // feat channel stride is HW_ (channels NOT contiguous per point) -> loop c,
  // loads stay coalesced across neighboring threads (same c, adjacent hw).
  const float* fb = feat + (long long)(bt * VC_ + vc) * C_ * HW_ + hw;
  for (int c = 0; c < C_; ++c) {
    const float f = fb[(long long)c * HW_];
    for (int k = 0; k < m; ++k)
      atomicAdd(&vf[(long long)lin8[k] * C_ + c], f * w8[k]);
  }
}

// ---------------------------------------------------------------------------
// Kernel 3: 1x1x1 conv as a WMMA GEMM, with the per-voxel normalization
// commuted into the epilogue (column scale commutes with the GEMM):
//   out[bt, o, v] = (1/max(vol_w[v],1e-6)) * sum_c proj_w[o,c]*vol_feat[v,c]
//                   + b[o]
// Block: 256 threads (8 waves). Each block handles a 16-voxel N-tile and all
// 256 output channels (16 M-tiles, 2 per wave). K = 256 as 64 steps of
// V_WMMA_F32_16X16X4_F32 (full f32 precision; phase is HBM-bandwidth bound).
// Tile staging is a pure copy -> use GLOBAL_LOAD_ASYNC_TO_LDS_B128 (ASYNCcnt)
// when available, avoiding the VGPR round-trip entirely.
//
// f32 16x16x4 fragment layout (ISA 7.12.2): lane = 16*h + m (h=0/1):
//   A: (v0,v1) = A[m][k0+2h], A[m][k0+2h+1]
//   B: (v0,v1) = B[k0+2h][n], B[k0+2h+1][n]
//   C/D vgpr r: M = r + 8*h, N = lane%16
// ---------------------------------------------------------------------------
__global__ void __launch_bounds__(256)
proj_gemm_wmma(const float* __restrict__ vol_feat,
               const float* __restrict__ vol_w,
               const float* __restrict__ proj_w,
               const float* __restrict__ proj_b,
               float* __restrict__ out)
{
  __shared__ float tile[16 * LDS_STRIDE];

  const int bt    = blockIdx.y;
  const int vbase = blockIdx.x * 16;
  const int tid   = threadIdx.x;

  // ---- stage a 16-voxel x 256-channel tile into LDS (coalesced copy) ----
  {
    const int r  = tid >> 4;            // voxel row 0..15
    const int cb = (tid & 15) * 16;     // 16 channels per thread (4 x b128)
    const float* src =
        vol_feat + ((long long)bt * NV_ + vbase + r) * C_ + cb;
    float* dst = &tile[r * LDS_STRIDE + cb];
#if ASYNC_STAGE
#pragma unroll
    for (int j = 0; j < 4; ++j) {
      __builtin_amdgcn_global_load_async_to_lds_b128(
          (__attribute__((address_space(1))) v4i_vs*)(src + 4 * j),
          (__attribute__((address_space(3))) v4i_vs*)(dst + 4 * j),
          /*offset=*/0, /*cpol=*/0);
    }
    asm volatile("s_wait_asynccnt 0" ::: "memory");
#else
#pragma unroll
    for (int j = 0; j < 4; ++j) {
      const float4 q = ((const float4*)src)[j];
      ((float4*)dst)[j] = q;
    }
#endif
  }
  __syncthreads();

  const int wave = tid >> 5;      // 0..7
  const int lane = tid & 31;
  const int half = lane >> 4;     // 0/1
  const int mn   = lane & 15;     // A row m / B col n
  const float* bsrc = &tile[mn * LDS_STRIDE + 2 * half];

  // Per-lane output-column normalization scale (commuted out of the GEMM).
  const float invw = 1.f / fmaxf(vol_w[bt * NV_ + vbase + mn], 1e-6f);

#pragma unroll
  for (int mt = 0; mt < 2; ++mt) {
    const int mbase = (wave * 2 + mt) * 16;
    const float* arow = proj_w + (mbase + mn) * C_ + 2 * half;
    v8f acc = {0.f, 0.f, 0.f, 0.f, 0.f, 0.f, 0.f, 0.f};
#pragma unroll 8
    for (int k0 = 0; k0 < C_; k0 += 4) {
      const v2f a = *(const v2f*)(arow + k0);   // global_load_b64 (L2-hot)
      const v2f b = *(const v2f*)(bsrc + k0);   // ds_load_b64, conflict-free
      acc = __builtin_amdgcn_wmma_f32_16x16x4_f32(
          /*neg_a=*/false, a, /*neg_b=*/false, b,
          /*c_mod=*/(short)0, acc, /*reuse_a=*/false, /*reuse_b=*/false);
    }
#pragma unroll
    for (int rr = 0; rr < 8; ++rr) {
      const int o = mbase + 8 * half + rr;
      out[((long long)bt * C_ + o) * NV_ + vbase + mn] =
          acc[rr] * invw + proj_b[o];
    }
  }
}

// ---------------------------------------------------------------------------
extern "C" void kernel_launch(void* const* d_in, const int* in_sizes, int n_in,
                              void* d_out, int out_size, void* d_ws, size_t ws_size,
                              hipStream_t stream) {
  const float* feat   = (const float*)d_in[0];  // (B,T,VC,C,H4,W4)
  const float* xyz    = (const float*)d_in[1];  // (B,T,VC,3,H4,W4)
  const float* conf   = (const float*)d_in[2];  // (B,T,VC,H4,W4)
  const float* proj_w = (const float*)d_in[3];  // (C,C)
  const float* proj_b = (const float*)d_in[4];  // (C,)
  float* out = (float*)d_out;                   // (B,T,C,NZ,NY,NX)

  // Workspace: vol_feat [BT*NV*C] f32 (204.8 MB) + vol_w [BT*NV] f32 (0.8 MB)
  float* vol_feat = (float*)d_ws;
  float* vol_w    = vol_feat + (size_t)BT_ * NV_ * C_;

  const long long nfloats = (long long)BT_ * NV_ * C_ + (long long)BT_ * NV_;
  zero_ws_kernel<<<dim3(8192), dim3(256), 0, stream>>>((float4*)d_ws, nfloats / 4);

  const int npts = BT_ * NPT_;  // 86016
  lift_scatter_kernel<<<dim3((npts + 255) / 256), dim3(256), 0, stream>>>(
      feat, xyz, conf, vol_feat, vol_w);

  proj_gemm_wmma<<<dim3(NV_ / 16, BT_), dim3(256), 0, stream>>>(
      vol_feat, vol_w, proj_w, proj_b, out);
}